// GINEE_12352325943908
// MI455X (gfx1250) — compile-verified
//
#include <hip/hip_runtime.h>

#define L_ 5
#define N_ 50000
#define E_ 600000
#define D_ 128
#define G_ 256
#define O_ 10
#define BN_EPS 1e-5f

typedef __attribute__((ext_vector_type(2))) float v2f;
typedef __attribute__((ext_vector_type(8))) float v8f;

__device__ __forceinline__ void atomic_add_f(float* p, float v) {
    __hip_atomic_fetch_add(p, v, __ATOMIC_RELAXED, __HIP_MEMORY_SCOPE_AGENT);
}

// ---------------------------------------------------------------------------
// Edge kernel: ea = sum_f bond_emb[f][attr[e,f]]; msg = relu(x[src]+ea);
// atomic scatter-add into aggr[dst]. 32 lanes per edge, float4 per lane.
// ---------------------------------------------------------------------------
__global__ __launch_bounds__(256)
void gine_scatter(const float* __restrict__ x, const int* __restrict__ ei,
                  const int* __restrict__ eattr, const float* __restrict__ emb,
                  float* __restrict__ aggr) {
    int tid = blockIdx.x * blockDim.x + threadIdx.x;
    int e = tid >> 5;
    if (e >= E_) return;
    int lane = tid & 31;
    int src = ei[e];
    int dst = ei[E_ + e];
    int a0 = eattr[e * 3 + 0];
    int a1 = eattr[e * 3 + 1];
    int a2 = eattr[e * 3 + 2];
    int c = lane * 4;
    float4 e0 = *(const float4*)(emb + (size_t)(a0) * D_ + c);
    float4 e1 = *(const float4*)(emb + (size_t)(8 + a1) * D_ + c);
    float4 e2 = *(const float4*)(emb + (size_t)(16 + a2) * D_ + c);
    float4 xv = *(const float4*)(x + (size_t)src * D_ + c);
    float4 msg = make_float4(
        fmaxf(xv.x + e0.x + e1.x + e2.x, 0.f),
        fmaxf(xv.y + e0.y + e1.y + e2.y, 0.f),
        fmaxf(xv.z + e0.z + e1.z + e2.z, 0.f),
        fmaxf(xv.w + e0.w + e1.w + e2.w, 0.f));
    float* ap = aggr + (size_t)dst * D_ + c;
    atomic_add_f(ap + 0, msg.x);
    atomic_add_f(ap + 1, msg.y);
    atomic_add_f(ap + 2, msg.z);
    atomic_add_f(ap + 3, msg.w);
}

// ---------------------------------------------------------------------------
// Node MLP: h = x + aggr; h = relu(BN1(h@W1+b1)); x' = relu(BN2(h@W2+b2))
// One wave per 16-row tile; WMMA f32 16x16x4 over K=128, 8 N-tiles, 2 GEMMs.
// ---------------------------------------------------------------------------
__global__ __launch_bounds__(64)
void node_mlp(const float* __restrict__ xin, const float* __restrict__ aggr,
              const float* __restrict__ W1, const float* __restrict__ B1,
              const float* __restrict__ G1, const float* __restrict__ Be1,
              const float* __restrict__ M1, const float* __restrict__ V1,
              const float* __restrict__ W2, const float* __restrict__ B2,
              const float* __restrict__ G2, const float* __restrict__ Be2,
              const float* __restrict__ M2, const float* __restrict__ V2,
              float* __restrict__ xout) {
    __shared__ float sIn[2][16][132];   // stride 132: conflict-free A reads
    __shared__ float sMid[2][16][132];
    const int wave = threadIdx.x >> 5;
    const int lane = threadIdx.x & 31;
    const int tile = blockIdx.x * 2 + wave;
    const int base = tile * 16;
    const int c4 = lane * 4;

    // Stage x + aggr tile into LDS (zeros for out-of-range rows).
    for (int r = 0; r < 16; ++r) {
        int node = base + r;                     // wave-uniform condition
        float4 v = make_float4(0.f, 0.f, 0.f, 0.f);
        if (node < N_) {
            float4 a = *(const float4*)(xin + (size_t)node * D_ + c4);
            float4 b = *(const float4*)(aggr + (size_t)node * D_ + c4);
            v = make_float4(a.x + b.x, a.y + b.y, a.z + b.z, a.w + b.w);
        }
        *(float4*)&sIn[wave][r][c4] = v;
    }
    __syncthreads();

    // WMMA f32 16x16x4 fragment mapping (ISA 7.12.2):
    // A: lane m=lane&15 holds rows; VGPR pair holds K = kb, kb+1 (kb = 0 or 2)
    const int m = lane & 15;
    const int kb = (lane >> 4) << 1;
    const int rowoff = (lane < 16) ? 0 : 8;      // C/D: vgpr r -> row r(+8)

    // ---- GEMM1 + BN1 + ReLU -> sMid ----
    for (int nt = 0; nt < 8; ++nt) {
        const int col = nt * 16 + m;
        v8f acc = {};
#pragma unroll
        for (int k0 = 0; k0 < D_; k0 += 4) {
            v2f a, b;
            a.x = sIn[wave][m][k0 + kb];
            a.y = sIn[wave][m][k0 + kb + 1];
            b.x = W1[(k0 + kb) * D_ + col];
            b.y = W1[(k0 + kb + 1) * D_ + col];
            acc = __builtin_amdgcn_wmma_f32_16x16x4_f32(
                false, a, false, b, (short)0, acc, false, false);
        }
        const float bias  = B1[col];
        const float scale = G1[col] * rsqrtf(V1[col] + BN_EPS);
        const float shift = Be1[col] - M1[col] * scale;
#pragma unroll
        for (int r = 0; r < 8; ++r) {
            float h = (acc[r] + bias) * scale + shift;
            sMid[wave][r + rowoff][col] = fmaxf(h, 0.f);
        }
    }
    __syncthreads();

    // ---- GEMM2 + BN2 + ReLU -> xout ----
    for (int nt = 0; nt < 8; ++nt) {
        const int col = nt * 16 + m;
        v8f acc = {};
#pragma unroll
        for (int k0 = 0; k0 < D_; k0 += 4) {
            v2f a, b;
            a.x = sMid[wave][m][k0 + kb];
            a.y = sMid[wave][m][k0 + kb + 1];
            b.x = W2[(k0 + kb) * D_ + col];
            b.y = W2[(k0 + kb + 1) * D_ + col];
            acc = __builtin_amdgcn_wmma_f32_16x16x4_f32(
                false, a, false, b, (short)0, acc, false, false);
        }
        const float bias  = B2[col];
        const float scale = G2[col] * rsqrtf(V2[col] + BN_EPS);
        const float shift = Be2[col] - M2[col] * scale;
#pragma unroll
        for (int r = 0; r < 8; ++r) {
            int node = base + r + rowoff;
            if (node < N_) {
                float h = (acc[r] + bias) * scale + shift;
                xout[(size_t)node * D_ + col] = fmaxf(h, 0.f);
            }
        }
    }
}

// ---------------------------------------------------------------------------
// Per-graph pooling (sum) and node counts.
// ---------------------------------------------------------------------------
__global__ __launch_bounds__(256)
void pool_sum(const float* __restrict__ h, const int* __restrict__ batch,
              float* __restrict__ pool) {
    int tid = blockIdx.x * blockDim.x + threadIdx.x;
    int node = tid >> 5;
    if (node >= N_) return;
    int lane = tid & 31;
    int g = batch[node];
    int c = lane * 4;
    float4 v = *(const float4*)(h + (size_t)node * D_ + c);
    float* p = pool + (size_t)g * D_ + c;
    atomic_add_f(p + 0, v.x);
    atomic_add_f(p + 1, v.y);
    atomic_add_f(p + 2, v.z);
    atomic_add_f(p + 3, v.w);
}

__global__ __launch_bounds__(256)
void node_count(const int* __restrict__ batch, float* __restrict__ counts) {
    int n = blockIdx.x * blockDim.x + threadIdx.x;
    if (n >= N_) return;
    atomic_add_f(&counts[batch[n]], 1.0f);
}

// ---------------------------------------------------------------------------
// Head: out[g,o] = sum_i (pool_i[g]/cnt[g]) @ fc_w[i][:,o] + sum_i fc_b[i][o]
// ---------------------------------------------------------------------------
__global__ __launch_bounds__(256)
void head_kernel(const float* __restrict__ pools, const float* __restrict__ counts,
                 const float* __restrict__ fcw, const float* __restrict__ fcb,
                 float* __restrict__ out) {
    int tid = blockIdx.x * blockDim.x + threadIdx.x;
    if (tid >= G_ * O_) return;
    int g = tid / O_;
    int o = tid - g * O_;
    float inv = 1.0f / fmaxf(counts[g], 1.0f);
    float acc = 0.f;
    for (int i = 0; i < L_ + 1; ++i) {
        const float* p = pools + ((size_t)i * G_ + g) * D_;
        const float* w = fcw + (size_t)i * D_ * O_ + o;
        float s = 0.f;
        for (int d = 0; d < D_; ++d) s += p[d] * w[d * O_];
        acc += s * inv + fcb[i * O_ + o];
    }
    out[(size_t)g * O_ + o] = acc;
}

// ---------------------------------------------------------------------------
extern "C" void kernel_launch(void* const* d_in, const int* in_sizes, int n_in,
                              void* d_out, int out_size, void* d_ws, size_t ws_size,
                              hipStream_t stream) {
    const float* x     = (const float*)d_in[0];
    const int*   ei    = (const int*)d_in[1];
    const int*   eattr = (const int*)d_in[2];
    const int*   batch = (const int*)d_in[3];
    const float* bemb  = (const float*)d_in[4];
    const float* w1    = (const float*)d_in[5];
    const float* b1    = (const float*)d_in[6];
    const float* cg    = (const float*)d_in[7];
    const float* cb    = (const float*)d_in[8];
    const float* cm    = (const float*)d_in[9];
    const float* cv    = (const float*)d_in[10];
    const float* w2    = (const float*)d_in[11];
    const float* b2    = (const float*)d_in[12];
    const float* bg    = (const float*)d_in[13];
    const float* bb    = (const float*)d_in[14];
    const float* bm    = (const float*)d_in[15];
    const float* bv    = (const float*)d_in[16];
    const float* fcw   = (const float*)d_in[17];
    const float* fcb   = (const float*)d_in[18];
    float* out = (float*)d_out;

    float* ws     = (float*)d_ws;
    float* xb0    = ws;
    float* xb1    = ws + (size_t)N_ * D_;
    float* aggr   = ws + 2 * (size_t)N_ * D_;
    float* pools  = ws + 3 * (size_t)N_ * D_;                  // (L+1)*G*D
    float* counts = pools + (size_t)(L_ + 1) * G_ * D_;        // G

    hipMemsetAsync(pools, 0,
                   ((size_t)(L_ + 1) * G_ * D_ + G_) * sizeof(float), stream);

    node_count<<<(N_ + 255) / 256, 256, 0, stream>>>(batch, counts);
    pool_sum<<<(N_ * 32 + 255) / 256, 256, 0, stream>>>(x, batch, pools);

    const int tiles = (N_ + 15) / 16;
    const int mlp_blocks = (tiles + 1) / 2;

    const float* xcur = x;
    float* xnext = xb0;
    for (int i = 0; i < L_; ++i) {
        hipMemsetAsync(aggr, 0, (size_t)N_ * D_ * sizeof(float), stream);
        gine_scatter<<<(E_ * 32 + 255) / 256, 256, 0, stream>>>(
            xcur, ei, eattr, bemb + (size_t)i * 3 * 8 * D_, aggr);
        node_mlp<<<mlp_blocks, 64, 0, stream>>>(
            xcur, aggr,
            w1 + (size_t)i * D_ * D_, b1 + (size_t)i * D_,
            cg + (size_t)i * D_, cb + (size_t)i * D_,
            cm + (size_t)i * D_, cv + (size_t)i * D_,
            w2 + (size_t)i * D_ * D_, b2 + (size_t)i * D_,
            bg + (size_t)i * D_, bb + (size_t)i * D_,
            bm + (size_t)i * D_, bv + (size_t)i * D_,
            xnext);
        pool_sum<<<(N_ * 32 + 255) / 256, 256, 0, stream>>>(
            xnext, batch, pools + (size_t)(i + 1) * G_ * D_);
        xcur = xnext;
        xnext = (xnext == xb0) ? xb1 : xb0;
    }

    head_kernel<<<(G_ * O_ + 255) / 256, 256, 0, stream>>>(
        pools, counts, fcw, fcb, out);
}